// GraphConstructorPang_12206297055831
// MI455X (gfx1250) — compile-verified
//
#include <hip/hip_runtime.h>

// GraphConstructorPang for MI455X (gfx1250), wave32 + f32 WMMA + async G->LDS.
//
// adj = tanh(3*(n1@n2^T - n2@n1^T)), n_i = tanh(3*(emb_i[idx]@W_i^T + b_i))
// then row-wise top-k(|adj + 0.01*noise|, k=64) mask applied in place on d_out.
//
// d_ws layout: n1 [8192*512] f32, then n2 [8192*512] f32  (32 MB total).

typedef float v2f __attribute__((ext_vector_type(2)));
typedef float v4f __attribute__((ext_vector_type(4)));
typedef float v8f __attribute__((ext_vector_type(8)));
typedef int   v4i __attribute__((ext_vector_type(4)));

#define N_NODES 8192
#define DIM     512
#define ALPHAC  3.0f

// Native V_TANH_F32 (confirmed lowering in round 3).
#if __has_builtin(__builtin_amdgcn_tanhf)
#define TANHF(x) __builtin_amdgcn_tanhf(x)
#elif __has_builtin(__builtin_amdgcn_tanh_f32)
#define TANHF(x) __builtin_amdgcn_tanh_f32(x)
#else
#define TANHF(x) tanhf(x)
#endif

// Async global->LDS (ASYNCcnt-tracked) staging, if available.
// Builtin parameter types are v4i* per hipcc diagnostics (AS1 src, AS3 dst).
#if __has_builtin(__builtin_amdgcn_global_load_async_to_lds_b128) && \
    __has_builtin(__builtin_amdgcn_s_wait_asynccnt)
#define USE_ASYNC_LDS 1
typedef __attribute__((address_space(1))) v4i* g_v4i_p;
typedef __attribute__((address_space(3))) v4i* l_v4i_p;
static __device__ __forceinline__ void async_g2l_b128(const float* g, float* l)
{
    __builtin_amdgcn_global_load_async_to_lds_b128(
        (g_v4i_p)(void*)const_cast<float*>(g), (l_v4i_p)(void*)l, 0, 0);
}
#else
#define USE_ASYNC_LDS 0
#endif

// ---------------------------------------------------------------------------
// Kernel 2 (placed first for disasm visibility):
// adj = tanh(ALPHA*(n1@n2^T - n2@n1^T)) -> d_out [8192x8192]
// Block tile 256(M)x128(N); 8 waves in a 4(M)x2(N) grid; each wave owns a
// 64x64 region = 4x4 tiles of 16x16 (128 accumulator VGPRs) -> 128 B of LDS
// fragment traffic per WMMA. K streamed in KC=16 chunks through DOUBLE-
// BUFFERED LDS staged with global_load_async_to_lds_b128: chunk c+1 is in
// flight while chunk c is consumed; s_wait_asynccnt(12) + in-order ASYNCcnt
// completion guarantees the current chunk has landed.
// The n1_J fragments are negated in registers to realize the subtraction.
// ---------------------------------------------------------------------------
#define BTM 256
#define BTN 128
#define KC  16
#define NCHUNK (DIM / KC)     // 32
#define LDP 20   // padded LDS row stride (20*4B = 80B, 16B-aligned rows)

__global__ __launch_bounds__(256)
void antisym_gemm_tanh(const float* __restrict__ n1,
                       const float* __restrict__ n2,
                       float* __restrict__ out)
{
    __shared__ float lA1[2][BTM * LDP];   // n1 rows of I-block (A, + term)
    __shared__ float lA2[2][BTM * LDP];   // n2 rows of I-block (A, - term)
    __shared__ float lB2[2][BTN * LDP];   // n2 rows of J-block (B, + term)
    __shared__ float lB1[2][BTN * LDP];   // n1 rows of J-block (B, - term)

    const int tid  = threadIdx.x;
    const int lane = tid & 31;
    const int wid  = tid >> 5;
    const int wm   = wid & 3;            // 0..3 : 64-row strip
    const int wn   = wid >> 2;           // 0..1 : 64-col strip
    const int iBase = blockIdx.y * BTM;
    const int jBase = blockIdx.x * BTN;

    const int r  = lane & 15;
    const int kh = (lane >> 4) * 2;

    // Stage one KC-chunk (12 b128 transfers per thread) into buffer `buf`.
    auto stage = [&](int kc, int buf) {
#if USE_ASYNC_LDS
#pragma unroll
        for (int t = 0; t < 4; ++t) {              // 1024 b128 per I-side matrix
            const int p   = tid + t * 256;
            const int row = p >> 2;
            const int c4  = (p & 3) * 4;
            const size_t gi = (size_t)(iBase + row) * DIM + kc + c4;
            async_g2l_b128(n1 + gi, lA1[buf] + row * LDP + c4);
            async_g2l_b128(n2 + gi, lA2[buf] + row * LDP + c4);
        }
#pragma unroll
        for (int t = 0; t < 2; ++t) {              // 512 b128 per J-side matrix
            const int p   = tid + t * 256;
            const int row = p >> 2;
            const int c4  = (p & 3) * 4;
            const size_t gj = (size_t)(jBase + row) * DIM + kc + c4;
            async_g2l_b128(n2 + gj, lB2[buf] + row * LDP + c4);
            async_g2l_b128(n1 + gj, lB1[buf] + row * LDP + c4);
        }
#else
#pragma unroll
        for (int t = 0; t < 4; ++t) {
            const int p   = tid + t * 256;
            const int row = p >> 2;
            const int c4  = (p & 3) * 4;
            const size_t gi = (size_t)(iBase + row) * DIM + kc + c4;
            *(v4f*)(lA1[buf] + row * LDP + c4) = *(const v4f*)(n1 + gi);
            *(v4f*)(lA2[buf] + row * LDP + c4) = *(const v4f*)(n2 + gi);
        }
#pragma unroll
        for (int t = 0; t < 2; ++t) {
            const int p   = tid + t * 256;
            const int row = p >> 2;
            const int c4  = (p & 3) * 4;
            const size_t gj = (size_t)(jBase + row) * DIM + kc + c4;
            *(v4f*)(lB2[buf] + row * LDP + c4) = *(const v4f*)(n2 + gj);
            *(v4f*)(lB1[buf] + row * LDP + c4) = *(const v4f*)(n1 + gj);
        }
#endif
    };

    v8f acc[4][4];
#pragma unroll
    for (int mt = 0; mt < 4; ++mt)
#pragma unroll
        for (int nt = 0; nt < 4; ++nt) acc[mt][nt] = 0.0f;

    stage(0, 0);                                   // prologue prefetch

    for (int c = 0; c < NCHUNK; ++c) {
        const int buf = c & 1;
        if (c + 1 < NCHUNK) {
            stage((c + 1) * KC, buf ^ 1);          // prefetch next chunk
#if USE_ASYNC_LDS
            __builtin_amdgcn_s_wait_asynccnt(12);  // chunk c landed (in-order)
#endif
        } else {
#if USE_ASYNC_LDS
            __builtin_amdgcn_s_wait_asynccnt(0);
#endif
        }
        __syncthreads();                           // chunk c visible to all waves

        // ---- 4 k-steps x 32 WMMA on buffer `buf`
#pragma unroll
        for (int k0 = 0; k0 < KC; k0 += 4) {
            v2f a1[4], a2[4], b2[4], b1[4];
#pragma unroll
            for (int mt = 0; mt < 4; ++mt) {
                const int m = wm * 64 + mt * 16 + r;
                a1[mt] = *(const v2f*)(lA1[buf] + m * LDP + k0 + kh);
                a2[mt] = *(const v2f*)(lA2[buf] + m * LDP + k0 + kh);
            }
#pragma unroll
            for (int nt = 0; nt < 4; ++nt) {
                const int cix = wn * 64 + nt * 16 + r;
                b2[nt] = *(const v2f*)(lB2[buf] + cix * LDP + k0 + kh);
                b1[nt] = -(*(const v2f*)(lB1[buf] + cix * LDP + k0 + kh));
            }
#pragma unroll
            for (int mt = 0; mt < 4; ++mt)
#pragma unroll
                for (int nt = 0; nt < 4; ++nt) {
                    acc[mt][nt] = __builtin_amdgcn_wmma_f32_16x16x4_f32(
                        false, a1[mt], false, b2[nt], (short)0, acc[mt][nt],
                        false, false);
                    acc[mt][nt] = __builtin_amdgcn_wmma_f32_16x16x4_f32(
                        false, a2[mt], false, b1[nt], (short)0, acc[mt][nt],
                        false, false);
                }
        }
        __syncthreads();   // all waves done reading `buf` before it is re-staged
    }

    // ---- Epilogue: adj = tanh(ALPHA * acc), native V_TANH_F32
    const int n    = lane & 15;
    const int moff = (lane >> 4) * 8;
#pragma unroll
    for (int mt = 0; mt < 4; ++mt)
#pragma unroll
        for (int nt = 0; nt < 4; ++nt) {
            const int colg = jBase + wn * 64 + nt * 16 + n;
#pragma unroll
            for (int v = 0; v < 8; ++v) {
                const int rowg = iBase + wm * 64 + mt * 16 + moff + v;
                out[(size_t)rowg * N_NODES + colg] =
                    TANHF(ALPHAC * acc[mt][nt][v]);
            }
        }
}

// ---------------------------------------------------------------------------
// Kernel 1: out[i][o] = tanh(ALPHA*(sum_k emb[idx[i]][k]*W[o][k] + b[o]))
// One wave per 16x16 output tile; K=512 via V_WMMA_F32_16X16X4_F32.
// tiles_m = 8192/16 = 512, tiles_n = 512/16 = 32 -> 16384 waves -> 2048 blocks.
// W (1 MB) and emb rows stay L2-resident (192 MB L2), so direct global frag
// loads are fine for this 8.6 GFLOP side computation.
// ---------------------------------------------------------------------------
__global__ __launch_bounds__(256)
void linear_tanh_wmma(const int* __restrict__ idx,
                      const float* __restrict__ emb,
                      const float* __restrict__ W,
                      const float* __restrict__ bias,
                      float* __restrict__ out)
{
    const int lane = threadIdx.x & 31;
    const int wid  = threadIdx.x >> 5;
    const int gw   = blockIdx.x * 8 + wid;   // global wave id
    const int tm   = gw >> 5;                // row tile (0..511)
    const int tn   = gw & 31;                // col tile (0..31)

    const int r  = lane & 15;                // A: M row / B: N col within tile
    const int kh = (lane >> 4) * 2;          // k-pair select (0 or 2)

    const float* __restrict__ arow = emb + (size_t)idx[tm * 16 + r] * DIM;
    const float* __restrict__ brow = W   + (size_t)(tn * 16 + r) * DIM;

    v8f acc = 0.0f;
#pragma unroll 8
    for (int k0 = 0; k0 < DIM; k0 += 4) {
        v2f a = *(const v2f*)(arow + k0 + kh);
        v2f b = *(const v2f*)(brow + k0 + kh);
        acc = __builtin_amdgcn_wmma_f32_16x16x4_f32(
                  false, a, false, b, (short)0, acc, false, false);
    }

    const int n    = lane & 15;
    const int moff = (lane >> 4) * 8;
    const float bn = bias[tn * 16 + n];
    float* __restrict__ o0 = out + (size_t)(tm * 16 + moff) * DIM + tn * 16 + n;
#pragma unroll
    for (int v = 0; v < 8; ++v)
        o0[(size_t)v * DIM] = TANHF(ALPHAC * (acc[v] + bn));
}

// ---------------------------------------------------------------------------
// Kernel 3: row-wise top-k mask, in place on d_out.
// key = |adj + 0.01*noise| >= 0, so IEEE bits are uint-monotone.
// Radix-descend binary search for the k-th largest key; keep key >= thr.
// One 256-thread block per row; 32 (key,val) pairs per thread in registers.
// ---------------------------------------------------------------------------
__global__ __launch_bounds__(256)
void topk_mask(float* __restrict__ out,
               const float* __restrict__ noise,
               const int* __restrict__ kptr)
{
    __shared__ unsigned scount;
    const int row = blockIdx.x;
    const int tid = threadIdx.x;
    const unsigned K = (unsigned)(*kptr);

    const float* __restrict__ nrow = noise + (size_t)row * N_NODES;
    float*       __restrict__ orow = out   + (size_t)row * N_NODES;

    float    val[32];
    unsigned key[32];
#pragma unroll
    for (int t = 0; t < 32; ++t) {
        const int j = tid + t * 256;
        const float v = orow[j];
        val[t] = v;
        key[t] = __float_as_uint(fabsf(v + 0.01f * nrow[j]));
    }

    unsigned thr = 0u;
    for (int bit = 30; bit >= 0; --bit) {
        const unsigned cand = thr | (1u << bit);
        if (tid == 0) scount = 0u;
        __syncthreads();
        unsigned local = 0u;
#pragma unroll
        for (int t = 0; t < 32; ++t) local += (key[t] >= cand) ? 1u : 0u;
        atomicAdd(&scount, local);
        __syncthreads();
        if (scount >= K) thr = cand;
        __syncthreads();   // all threads read scount before next reset
    }

#pragma unroll
    for (int t = 0; t < 32; ++t) {
        const int j = tid + t * 256;
        orow[j] = (key[t] >= thr) ? val[t] : 0.0f;
    }
}

// ---------------------------------------------------------------------------
extern "C" void kernel_launch(void* const* d_in, const int* in_sizes, int n_in,
                              void* d_out, int out_size, void* d_ws, size_t ws_size,
                              hipStream_t stream)
{
    (void)in_sizes; (void)n_in; (void)out_size; (void)ws_size;
    const int*   idx   = (const int*)  d_in[0];
    const float* emb1  = (const float*)d_in[1];
    const float* emb2  = (const float*)d_in[2];
    const float* w1    = (const float*)d_in[3];
    const float* b1    = (const float*)d_in[4];
    const float* w2    = (const float*)d_in[5];
    const float* b2    = (const float*)d_in[6];
    const float* noise = (const float*)d_in[7];
    const int*   kptr  = (const int*)  d_in[8];
    float* out = (float*)d_out;

    float* n1 = (float*)d_ws;                       // 8192*512 f32 (16 MB)
    float* n2 = n1 + (size_t)N_NODES * DIM;         // 8192*512 f32 (16 MB)

    const dim3 blk(256);
    linear_tanh_wmma<<<dim3(2048), blk, 0, stream>>>(idx, emb1, w1, b1, n1);
    linear_tanh_wmma<<<dim3(2048), blk, 0, stream>>>(idx, emb2, w2, b2, n2);
    antisym_gemm_tanh<<<dim3(N_NODES / BTN, N_NODES / BTM), blk, 0, stream>>>(n1, n2, out);
    topk_mask<<<dim3(N_NODES), blk, 0, stream>>>(out, noise, kptr);
}